// DigitCaps_69114613730131
// MI455X (gfx1250) — compile-verified
//
#include <hip/hip_runtime.h>
#include <hip/hip_bf16.h>
#include <math.h>

// DigitCaps (dead-routing-faithful): out[j*16+d] = squash_elem( (1/512) * sum_{i,k} W[i,j,d,k]*x[i,k] )
// Memory-bound GEMV (2.62 MB of W, 1.3 MFLOP). Contraction executed on the CDNA5
// matrix pipe via V_WMMA_F32_16X16X4_F32 (A = 16x4 W-tile, B = x broadcast over N).

typedef __attribute__((ext_vector_type(2))) float v2f;
typedef __attribute__((ext_vector_type(8))) float v8f;

#define N_IN  512
#define N_OUT 10
#define D_OUT 16
#define D_IN  8
#define CAPS_EPS 1e-7f

#define BLOCKS 32
#define WAVES_PER_BLOCK 8
#define TOTAL_WAVES (BLOCKS * WAVES_PER_BLOCK)   // 256 waves
#define I_PER_WAVE (N_IN / TOTAL_WAVES)          // 2 primary caps per wave
#define M_TOTAL (N_OUT * D_OUT)                  // 160 outputs

__global__ void __launch_bounds__(WAVES_PER_BLOCK * 32)
caps_partial_kernel(const float* __restrict__ x,
                    const float* __restrict__ W,
                    float* __restrict__ ws) {
    const int lane        = threadIdx.x & 31;
    const int waveInBlock = threadIdx.x >> 5;
    const int wave        = blockIdx.x * WAVES_PER_BLOCK + waveInBlock;
    const int laneHi      = lane >> 4;    // 0: K=0/1 half, 1: K=2/3 half
    const int mrow        = lane & 15;    // A-tile row (d index)

    v8f acc[N_OUT];
#pragma unroll
    for (int j = 0; j < N_OUT; ++j) acc[j] = (v8f){0.f,0.f,0.f,0.f,0.f,0.f,0.f,0.f};

    const int i0 = wave * I_PER_WAVE;
    for (int ii = 0; ii < I_PER_WAVE; ++ii) {
        const int i = i0 + ii;

        // B tiles: x[i, 0:8] replicated across all 16 N-columns.
        // B 4x16 f32 layout: v0 lanes0-15 = K0, lanes16-31 = K2; v1 = K1 / K3.
        const float* __restrict__ xi = x + (size_t)i * D_IN;
        v2f b0, b1;
        b0.x = xi[0 + 2 * laneHi];
        b0.y = xi[1 + 2 * laneHi];
        b1.x = xi[4 + 2 * laneHi];
        b1.y = xi[5 + 2 * laneHi];

        const float* __restrict__ Wi = W + (size_t)i * (N_OUT * D_OUT * D_IN);
#pragma unroll
        for (int j = 0; j < N_OUT; ++j) {
            // A 16x4 f32 layout: lane m holds (M=m, K=0..1) in v0/v1; lane m+16 holds K=2..3.
            const float* __restrict__ Wj =
                Wi + j * (D_OUT * D_IN) + mrow * D_IN + 2 * laneHi;
            v2f a0, a1;
            a0.x = Wj[0]; a0.y = Wj[1];       // k = 0..3 tile
            a1.x = Wj[4]; a1.y = Wj[5];       // k = 4..7 tile

            acc[j] = __builtin_amdgcn_wmma_f32_16x16x4_f32(
                false, a0, false, b0, (short)0, acc[j], false, false);
            acc[j] = __builtin_amdgcn_wmma_f32_16x16x4_f32(
                false, a1, false, b1, (short)0, acc[j], false, false);
        }
    }

    // All B-columns are identical; extract column 0.
    // C/D layout: VGPR r, lanes0-15 -> M=r; lanes16-31 -> M=8+r.
    if (mrow == 0) {
        float* __restrict__ dst = ws + (size_t)wave * M_TOTAL + laneHi * 8;
#pragma unroll
        for (int j = 0; j < N_OUT; ++j) {
#pragma unroll
            for (int r = 0; r < 8; ++r) {
                dst[j * D_OUT + r] = acc[j][r];
            }
        }
    }
}

__global__ void caps_finalize_kernel(const float* __restrict__ ws,
                                     float* __restrict__ out) {
    const int m = threadIdx.x;
    if (m >= M_TOTAL) return;
    float s = 0.f;
    for (int w = 0; w < TOTAL_WAVES; ++w) {
        s += ws[(size_t)w * M_TOTAL + m];
    }
    s *= (1.0f / (float)N_IN);            // cij = 1/512
    const float sq   = s * s;             // per-element "norm" (last axis size 1)
    const float norm = s / (sqrtf(sq + CAPS_EPS) + CAPS_EPS);
    out[m] = (sq / (1.0f + sq)) * norm;
}

extern "C" void kernel_launch(void* const* d_in, const int* in_sizes, int n_in,
                              void* d_out, int out_size, void* d_ws, size_t ws_size,
                              hipStream_t stream) {
    const float* x = (const float*)d_in[0];   // (512, 8) fp32
    const float* W = (const float*)d_in[1];   // (1, 512, 10, 16, 8) fp32
    float* out = (float*)d_out;               // 160 fp32
    float* ws  = (float*)d_ws;                // 256 * 160 fp32 partials (160 KB)

    caps_partial_kernel<<<BLOCKS, WAVES_PER_BLOCK * 32, 0, stream>>>(x, W, ws);
    caps_finalize_kernel<<<1, 192, 0, stream>>>(ws, out);
}